// NiDAQ_NET_78271484003268
// MI455X (gfx1250) — compile-verified
//
#include <hip/hip_runtime.h>
#include <hip/hip_bf16.h>

typedef __attribute__((ext_vector_type(2))) float v2f;
typedef __attribute__((ext_vector_type(8))) float v8f;

#define BATCH   256
#define NIN     784
#define NH      100
#define NELEM   (BATCH * NH)      // 25600 state elements
#define TSTEPS  2500
#define NCHUNK  5
#define CHUNK   (TSTEPS / NCHUNK) // 500

// ---------------------------------------------------------------------------
// Kernel 1: cur1 = x @ W1^T + b1 via V_WMMA_F32_16X16X4_F32.
// One wave32 computes one 16x16 output tile; K-loop of 784/4 = 196 WMMAs.
// A (16x4, M=x-rows): lanes 0-15 hold M=0..15 / K={0,1}; lanes 16-31 M=0..15 / K={2,3}.
// B (4x16,  N=W1-rows): mirrored layout with N in place of M.
// C/D (16x16 f32): VGPR i -> M = i + 8*(lane>=16), N = lane&15.
//
// Out-of-range N lanes (n >= 100) load a CLAMPED W1 row instead of branching:
// B column n only feeds D column n, and D columns >= 100 are never stored,
// so the inner loop is fully branch-free (EXEC stays all-1s for WMMA).
// ---------------------------------------------------------------------------
__global__ __launch_bounds__(32) void snn_gemm_wmma(
    const float* __restrict__ x,    // [BATCH, NIN]
    const float* __restrict__ W1,   // [NH, NIN]
    const float* __restrict__ b1,   // [NH]
    float* __restrict__ cur1)       // [BATCH, NH]
{
    const int lane  = threadIdx.x;        // 0..31
    const int half  = lane >> 4;          // 0 or 1 (K-half)
    const int l     = lane & 15;
    const int tileM = blockIdx.x;         // 0..15  (256/16)
    const int tileN = blockIdx.y;         // 0..6   (ceil(100/16))

    const int m  = tileM * 16 + l;        // x row for A-fragment loads
    const int n  = tileN * 16 + l;        // W1 row (output column) for B loads
    const int nc = (n < NH) ? n : (NH - 1); // clamped: garbage only hits dead columns

    const float* xrow = x  + (size_t)m  * NIN;
    const float* wrow = W1 + (size_t)nc * NIN;

    v8f acc = {};
    #pragma unroll 4
    for (int k = 0; k < NIN; k += 4) {
        const int ka = k + half * 2;      // this lane's K pair
        // 8-byte vector loads (rows are 8B aligned: 784*4 = 3136 = 8*392)
        const float2 av = *(const float2*)(xrow + ka);
        const float2 bv = *(const float2*)(wrow + ka);
        v2f a; a.x = av.x; a.y = av.y;
        v2f b; b.x = bv.x; b.y = bv.y;
        // 8 args: (neg_a, A, neg_b, B, c_mod, C, reuse_a, reuse_b)
        acc = __builtin_amdgcn_wmma_f32_16x16x4_f32(
            /*neg_a=*/false, a, /*neg_b=*/false, b,
            /*c_mod=*/(short)0, acc, /*reuse_a=*/false, /*reuse_b=*/false);
    }

    if (n < NH) {
        const float bias = b1[n];
        #pragma unroll
        for (int i = 0; i < 8; ++i) {
            const int row = tileM * 16 + half * 8 + i;
            cur1[(size_t)row * NH + n] = acc[i] + bias;
        }
    }
}

// ---------------------------------------------------------------------------
// Kernel 2: the 2500-step LIF recurrence + trajectory recording.
// Store-bandwidth bound (1.28 GB of output -> ~55 us floor at 23.3 TB/s).
// One thread per (b,h) element; blockIdx.y picks a time chunk: replay the
// prefix (compute only, ~free VALU), then record CHUNK steps with coalesced
// non-temporal streaming stores.
// ---------------------------------------------------------------------------
__global__ __launch_bounds__(256) void snn_recurrence(
    const float* __restrict__ cur1,     // [NELEM]
    const float* __restrict__ alpha1_p, // scalar
    const float* __restrict__ beta1_p,  // scalar
    float* __restrict__ out)            // 5 x [TSTEPS, NELEM] concatenated
{
    const int idx = blockIdx.x * blockDim.x + threadIdx.x;
    if (idx >= NELEM) return;

    const float a1  = fminf(fmaxf(alpha1_p[0], 0.0f), 1.0f);
    const float be1 = fminf(fmaxf(beta1_p[0],  0.0f), 1.0f);
    const float cur = cur1[idx];

    const size_t field = (size_t)TSTEPS * NELEM;
    float* __restrict__ spk2_o = out;
    float* __restrict__ mem2_o = out + field;
    float* __restrict__ cur_o  = out + 2 * field;
    float* __restrict__ spk1_o = out + 3 * field;
    float* __restrict__ mem1_o = out + 4 * field;

    float syn1 = 0.0f, mem1 = 0.0f, syn2 = 0.0f, mem2 = 0.0f;

    const int t0 = blockIdx.y * CHUNK;
    const int t1 = t0 + CHUNK;

    // Warm-up replay of the prefix (no stores): pure VALU, hidden by other
    // chunks' store traffic.
    for (int t = 0; t < t0; ++t) {
        const float reset1 = ((mem1 - 1.0f) > 0.0f) ? 1.0f : 0.0f;
        syn1 = a1 * syn1 + cur;
        mem1 = (be1 * mem1 + syn1) * (1.0f - reset1);
        const float spk1 = ((mem1 - 1.0f) > 0.0f) ? 1.0f : 0.0f;
        const float reset2 = ((mem2 - 1.0f) > 0.0f) ? 1.0f : 0.0f;
        syn2 = 0.8f * syn2 + spk1;
        mem2 = (0.9f * mem2 + syn2) * (1.0f - reset2);
    }

    // Recorded window: 5 coalesced streaming stores per step.
    for (int t = t0; t < t1; ++t) {
        const float reset1 = ((mem1 - 1.0f) > 0.0f) ? 1.0f : 0.0f;
        syn1 = a1 * syn1 + cur;
        mem1 = (be1 * mem1 + syn1) * (1.0f - reset1);
        const float spk1 = ((mem1 - 1.0f) > 0.0f) ? 1.0f : 0.0f;
        const float reset2 = ((mem2 - 1.0f) > 0.0f) ? 1.0f : 0.0f;
        syn2 = 0.8f * syn2 + spk1;
        mem2 = (0.9f * mem2 + syn2) * (1.0f - reset2);
        const float spk2 = ((mem2 - 1.0f) > 0.0f) ? 1.0f : 0.0f;

        const size_t off = (size_t)t * NELEM + idx;
        __builtin_nontemporal_store(spk2, spk2_o + off);
        __builtin_nontemporal_store(mem2, mem2_o + off);
        __builtin_nontemporal_store(cur,  cur_o  + off);
        __builtin_nontemporal_store(spk1, spk1_o + off);
        __builtin_nontemporal_store(mem1, mem1_o + off);
    }
}

extern "C" void kernel_launch(void* const* d_in, const int* in_sizes, int n_in,
                              void* d_out, int out_size, void* d_ws, size_t ws_size,
                              hipStream_t stream) {
    const float* x      = (const float*)d_in[0];  // [256, 784]
    const float* W1     = (const float*)d_in[1];  // [100, 784]
    const float* b1     = (const float*)d_in[2];  // [100]
    const float* alpha1 = (const float*)d_in[3];  // scalar
    const float* beta1  = (const float*)d_in[4];  // scalar
    float* out  = (float*)d_out;
    float* cur1 = (float*)d_ws;                   // 25600 floats of scratch

    // GEMM: 16 M-tiles x 7 N-tiles, one wave32 per tile.
    dim3 ggrid(BATCH / 16, (NH + 15) / 16);
    snn_gemm_wmma<<<ggrid, 32, 0, stream>>>(x, W1, b1, cur1);

    // Recurrence: 100 blocks of 256 threads over elements x 5 time chunks.
    dim3 sgrid((NELEM + 255) / 256, NCHUNK);
    snn_recurrence<<<sgrid, 256, 0, stream>>>(cur1, alpha1, beta1, out);
}